// SpatiallySparse_Dense_17128329576939
// MI455X (gfx1250) — compile-verified
//
#include <hip/hip_runtime.h>
#include <hip/hip_bf16.h>
#include <stddef.h>

// ---------------------------------------------------------------------------
// Problem constants (from the reference): B=16, N=32*32*32=32768, L=64, f32.
// ---------------------------------------------------------------------------
#define NVOX 32768
#define LDIM 64
#define NBATCH 16
#define CK 32        // K-chunk staged through LDS per iteration
#define KS_MTM 32    // split-K factor for S = M^T M      (128 blocks)
#define KS_G   16    // split-K factor for masked grams   (1024 blocks)

typedef __attribute__((ext_vector_type(2))) float v2f;
typedef __attribute__((ext_vector_type(8))) float v8f;

// V_WMMA_F32_16X16X4_F32: D(16x16 f32) = A(16x4 f32) * B(4x16 f32) + C
// per-lane: A,B = 2 VGPRs (v2f), C/D = 8 VGPRs (v8f).
__device__ __forceinline__ v8f wmma4(v2f a, v2f b, v8f c) {
  return __builtin_amdgcn_wmma_f32_16x16x4_f32(
      /*neg_a=*/false, a, /*neg_b=*/false, b,
      /*c_mod=*/(short)0, c, /*reuse_a=*/false, /*reuse_b=*/false);
}

// ---------------------------------------------------------------------------
// Pass 1 (split-K): partial S = M^T M over a slice of N.
// grid (2, 2, KS_MTM); one wave per 32x32 output tile slice.
// LDS tiles transposed: lA[l_local][n_local] = M[n, abase+l]; stride 34 makes
// the v2f operand reads cover 62 of 64 banks (lanes 0 and 31 share a bank pair).
// ---------------------------------------------------------------------------
__global__ __launch_bounds__(32) void k_gram_mtm_part(const float* __restrict__ M,
                                                      float* __restrict__ Spart) {
  __shared__ __align__(16) float lA[32][34];
  __shared__ __align__(16) float lB[32][34];
  const int lane  = threadIdx.x;
  const int abase = blockIdx.x * 32;   // output rows  (l)
  const int bbase = blockIdx.y * 32;   // output cols  (k)
  const int kz    = blockIdx.z;
  const int cbeg  = kz * (NVOX / KS_MTM);
  const int cend  = cbeg + (NVOX / KS_MTM);
  const int r = lane & 15;
  const int h = lane >> 4;

  v8f acc00 = {}, acc01 = {}, acc10 = {}, acc11 = {};

  for (int c0 = cbeg; c0 < cend; c0 += CK) {
    // Stage (transposing): 16 consecutive floats per half-wave -> coalesced.
    for (int idx = lane; idx < 32 * 32; idx += 32) {
      const int n = idx >> 5, col = idx & 31;
      lA[col][n] = M[(size_t)(c0 + n) * LDIM + abase + col];
      lB[col][n] = M[(size_t)(c0 + n) * LDIM + bbase + col];
    }
    __syncthreads();
#pragma unroll
    for (int s = 0; s < CK / 4; ++s) {
      const int nl = 4 * s + 2 * h;            // even -> 8B aligned
      v2f a0 = *(const v2f*)&lA[r][nl];
      v2f a1 = *(const v2f*)&lA[r + 16][nl];
      v2f b0 = *(const v2f*)&lB[r][nl];
      v2f b1 = *(const v2f*)&lB[r + 16][nl];
      acc00 = wmma4(a0, b0, acc00);
      acc01 = wmma4(a0, b1, acc01);
      acc10 = wmma4(a1, b0, acc10);
      acc11 = wmma4(a1, b1, acc11);
    }
    __syncthreads();
  }
  // C/D layout: VGPR i -> row i (lanes 0-15) / i+8 (lanes 16-31), col = lane&15
  float* __restrict__ Sp = Spart + (size_t)kz * (LDIM * LDIM);
#pragma unroll
  for (int i = 0; i < 8; ++i) {
    const int row0 = abase + i + 8 * h;
    Sp[(size_t)(row0)      * LDIM + bbase + r]      = acc00[i];
    Sp[(size_t)(row0)      * LDIM + bbase + 16 + r] = acc01[i];
    Sp[(size_t)(row0 + 16) * LDIM + bbase + r]      = acc10[i];
    Sp[(size_t)(row0 + 16) * LDIM + bbase + 16 + r] = acc11[i];
  }
}

// ---------------------------------------------------------------------------
// Generic deterministic partial-sum reduction: dst[j] = sum_k part[k*elems+j].
// Ascending k -> fixed FP order -> bitwise deterministic.
// ---------------------------------------------------------------------------
__global__ __launch_bounds__(256) void k_reduce_parts(const float* __restrict__ part,
                                                      float* __restrict__ dst,
                                                      int nparts, int elems) {
  const int j = blockIdx.x * 256 + threadIdx.x;
  if (j >= elems) return;
  float acc = 0.0f;
  for (int k = 0; k < nparts; ++k) acc += part[(size_t)k * elems + j];
  dst[j] = acc;
}

// ---------------------------------------------------------------------------
// Pass 2: Sinv = inv(S) via Gauss-Jordan (S is SPD -> diagonal pivots OK).
// One block, 64 threads; thread r owns row r of the augmented [S | I].
// ---------------------------------------------------------------------------
__global__ __launch_bounds__(64) void k_inv64(const float* __restrict__ S,
                                              float* __restrict__ Sinv) {
  __shared__ float A[64][129];
  const int r = threadIdx.x;
  for (int c = 0; c < 64; ++c) {
    A[r][c]      = S[r * 64 + c];
    A[r][64 + c] = (r == c) ? 1.0f : 0.0f;
  }
  for (int p = 0; p < 64; ++p) {
    __syncthreads();
    if (r != p) {
      const float f = A[r][p] / A[p][p];
      for (int c = p; c < 128; ++c) A[r][c] -= f * A[p][c];
    }
  }
  __syncthreads();
  const float d = A[r][r];
  for (int c = 0; c < 64; ++c) Sinv[r * 64 + c] = A[r][64 + c] / d;
}

// ---------------------------------------------------------------------------
// Pass 3: W = Sinv @ M^T  -> W[l][n], row-major [64][NVOX].
// grid (NVOX/64, 16), block 256: thread computes W[lb+lq][nb+nl].
// ---------------------------------------------------------------------------
__global__ __launch_bounds__(256) void k_make_w(const float* __restrict__ M,
                                                const float* __restrict__ Sinv,
                                                float* __restrict__ W) {
  __shared__ float lM[64][65];    // padded: lanes read [n][k] conflict-free
  __shared__ float lS[4][64];
  const int tid = threadIdx.x;
  const int nb  = blockIdx.x * 64;
  const int lb  = blockIdx.y * 4;

  lS[tid >> 6][tid & 63] = Sinv[(lb + (tid >> 6)) * 64 + (tid & 63)];
  for (int idx = tid; idx < 64 * 64; idx += 256)
    lM[idx >> 6][idx & 63] = M[(size_t)(nb + (idx >> 6)) * LDIM + (idx & 63)];
  __syncthreads();

  const int nl = tid & 63, lq = tid >> 6;
  float acc = 0.0f;
#pragma unroll 8
  for (int k = 0; k < 64; ++k) acc += lS[lq][k] * lM[nl][k];
  W[(size_t)(lb + lq) * NVOX + nb + nl] = acc;   // 64 consecutive -> coalesced
}

// ---------------------------------------------------------------------------
// Pass 4 (dominant, split-K): partial G_b = (W.*m_b)(W.*m_b)^T over an
// N-slice.  grid (2, 2, NBATCH*KS_G) = 1024 single-wave blocks.
// Mask applied while staging into LDS; 4 WMMA tiles per wave for reuse.
// ---------------------------------------------------------------------------
__global__ __launch_bounds__(32) void k_gram_masked_part(const float* __restrict__ W,
                                                         const float* __restrict__ mask,
                                                         float* __restrict__ Gpart) {
  __shared__ __align__(16) float lA[32][34];
  __shared__ __align__(16) float lB[32][34];
  const int lane  = threadIdx.x;
  const int abase = blockIdx.x * 32;
  const int bbase = blockIdx.y * 32;
  const int b     = blockIdx.z & (NBATCH - 1);
  const int kz    = blockIdx.z >> 4;
  const int cbeg  = kz * (NVOX / KS_G);
  const int cend  = cbeg + (NVOX / KS_G);
  const float* __restrict__ mrow = mask + (size_t)b * NVOX;
  const int r = lane & 15;
  const int h = lane >> 4;

  v8f acc00 = {}, acc01 = {}, acc10 = {}, acc11 = {};

  for (int c0 = cbeg; c0 < cend; c0 += CK) {
    for (int idx = lane; idx < 32 * 32; idx += 32) {
      const int rr = idx >> 5, cc = idx & 31;     // cc consecutive -> coalesced
      const float mv = mrow[c0 + cc];
      lA[rr][cc] = W[(size_t)(abase + rr) * NVOX + c0 + cc] * mv;
      lB[rr][cc] = W[(size_t)(bbase + rr) * NVOX + c0 + cc] * mv;
    }
    __syncthreads();
#pragma unroll
    for (int s = 0; s < CK / 4; ++s) {
      const int nl = 4 * s + 2 * h;
      v2f a0 = *(const v2f*)&lA[r][nl];
      v2f a1 = *(const v2f*)&lA[r + 16][nl];
      v2f b0 = *(const v2f*)&lB[r][nl];
      v2f b1 = *(const v2f*)&lB[r + 16][nl];
      acc00 = wmma4(a0, b0, acc00);
      acc01 = wmma4(a0, b1, acc01);
      acc10 = wmma4(a1, b0, acc10);
      acc11 = wmma4(a1, b1, acc11);
    }
    __syncthreads();
  }
  // Partial for (kz, b): layout Gpart[(kz*NBATCH + b)][64][64]
  float* __restrict__ Gp = Gpart + (size_t)(kz * NBATCH + b) * (LDIM * LDIM);
#pragma unroll
  for (int i = 0; i < 8; ++i) {
    const int row0 = abase + i + 8 * h;
    Gp[(size_t)(row0)      * LDIM + bbase + r]      = acc00[i];
    Gp[(size_t)(row0)      * LDIM + bbase + 16 + r] = acc01[i];
    Gp[(size_t)(row0 + 16) * LDIM + bbase + r]      = acc10[i];
    Gp[(size_t)(row0 + 16) * LDIM + bbase + 16 + r] = acc11[i];
  }
}

// ---------------------------------------------------------------------------
// Pass 5: t[b][l] = sum_n W[l,n]*mask[b,n]*y[b,n].  grid (64,16), block 256.
// ---------------------------------------------------------------------------
__global__ __launch_bounds__(256) void k_woty(const float* __restrict__ W,
                                              const float* __restrict__ mask,
                                              const float* __restrict__ y,
                                              float* __restrict__ t) {
  __shared__ float red[256];
  const int l = blockIdx.x, b = blockIdx.y, tid = threadIdx.x;
  const float* __restrict__ wr = W    + (size_t)l * NVOX;
  const float* __restrict__ mr = mask + (size_t)b * NVOX;
  const float* __restrict__ yr = y    + (size_t)b * NVOX;
  float acc = 0.0f;
  for (int n = tid; n < NVOX; n += 256) acc += wr[n] * mr[n] * yr[n];
  red[tid] = acc;
  __syncthreads();
  for (int s = 128; s > 0; s >>= 1) {
    if (tid < s) red[tid] += red[tid + s];
    __syncthreads();
  }
  if (tid == 0) t[b * LDIM + l] = red[0];
}

// ---------------------------------------------------------------------------
// Pass 6: per-batch solve G_b x = t_b (== inv(G_b) @ t_b) via Gauss-Jordan.
// grid 16, block 64; thread r owns row r.  Writes d_out directly.
// ---------------------------------------------------------------------------
__global__ __launch_bounds__(64) void k_solve(const float* __restrict__ G,
                                              const float* __restrict__ t,
                                              float* __restrict__ out) {
  __shared__ float A[64][65];
  __shared__ float v[64];
  const int r = threadIdx.x, b = blockIdx.x;
  const float* __restrict__ Gb = G + (size_t)b * (LDIM * LDIM);
  for (int c = 0; c < 64; ++c) A[r][c] = Gb[r * 64 + c];
  v[r] = t[b * LDIM + r];
  for (int p = 0; p < 64; ++p) {
    __syncthreads();
    if (r != p) {
      const float f = A[r][p] / A[p][p];
      for (int c = p; c < 64; ++c) A[r][c] -= f * A[p][c];
      v[r] -= f * v[p];
    }
  }
  __syncthreads();
  out[b * LDIM + r] = v[r] / A[r][r];
}

// ---------------------------------------------------------------------------
// Launch.  Workspace layout (floats), total ~13.4 MB:
//   S:4096 | Sinv:4096 | t:1024 | pad | G:65536 | Spart:131072 |
//   Gpart:1048576 | W:2097152
// ---------------------------------------------------------------------------
extern "C" void kernel_launch(void* const* d_in, const int* in_sizes, int n_in,
                              void* d_out, int out_size, void* d_ws, size_t ws_size,
                              hipStream_t stream) {
  (void)in_sizes; (void)n_in; (void)out_size; (void)ws_size;
  const float* y    = (const float*)d_in[0];   // [16, 32768]
  const float* mask = (const float*)d_in[1];   // [16, 32768]
  const float* M    = (const float*)d_in[2];   // [32768, 64]
  float* out = (float*)d_out;                  // [16, 64]

  float* ws    = (float*)d_ws;
  float* S     = ws;                     // 64*64
  float* Sinv  = ws + 4096;              // 64*64
  float* t     = ws + 8192;              // 16*64
  float* G     = ws + 16384;             // 16*64*64              = 65536
  float* Spart = ws + 81920;             // KS_MTM*64*64          = 131072
  float* Gpart = ws + 212992;            // KS_G*16*64*64         = 1048576
  float* W     = ws + 1261568;           // 64*32768              = 2097152

  k_gram_mtm_part   <<<dim3(2, 2, KS_MTM),          32, 0, stream>>>(M, Spart);
  k_reduce_parts    <<<(LDIM * LDIM) / 256,        256, 0, stream>>>(Spart, S, KS_MTM, LDIM * LDIM);
  k_inv64           <<<1,                           64, 0, stream>>>(S, Sinv);
  k_make_w          <<<dim3(NVOX / 64, 16),        256, 0, stream>>>(M, Sinv, W);
  k_gram_masked_part<<<dim3(2, 2, NBATCH * KS_G),   32, 0, stream>>>(W, mask, Gpart);
  k_reduce_parts    <<<(NBATCH * LDIM * LDIM) / 256, 256, 0, stream>>>(Gpart, G, KS_G, NBATCH * LDIM * LDIM);
  k_woty            <<<dim3(64, NBATCH),           256, 0, stream>>>(W, mask, y, t);
  k_solve           <<<NBATCH,                      64, 0, stream>>>(G, t, out);
}